// AttentionRNNModel_78305843741050
// MI455X (gfx1250) — compile-verified
//
#include <hip/hip_runtime.h>
#include <hip/hip_bf16.h>

// Problem sizes (match reference)
#define B_  32
#define T_  128
#define V_  32000
#define E_  128
#define H_  256

typedef __attribute__((ext_vector_type(2))) float v2f;
typedef __attribute__((ext_vector_type(8))) float v8f;
typedef __attribute__((ext_vector_type(4))) int   v4i;

// D = A(16x4) * B(4x16) + C, fp32, wave32 WMMA
#define WMMA_F32_16x16x4(a, b, c) \
  __builtin_amdgcn_wmma_f32_16x16x4_f32(false, (a), false, (b), (short)0, (c), false, false)

// ---------------------------------------------------------------------------
// CDNA5 async global->LDS path (ASYNCcnt-tracked DMA), with sync fallback.
// Probe-derived prototype: (int4* src_global, int4* dst_lds, int imm, int cpol)
// ---------------------------------------------------------------------------
#if defined(__HIP_DEVICE_COMPILE__) && __has_builtin(__builtin_amdgcn_global_load_async_to_lds_b128)
#define HAVE_ASYNC_LDS 1
#else
#define HAVE_ASYNC_LDS 0
#endif

#if HAVE_ASYNC_LDS
#if __has_builtin(__builtin_amdgcn_s_wait_asynccnt)
#define WAIT_ASYNC(n) __builtin_amdgcn_s_wait_asynccnt(n)
#else
#define WAIT_ASYNC(n) asm volatile("s_wait_asynccnt %0" :: "i"(n) : "memory")
#endif
#define ASYNC_CP16(dst, src) \
  __builtin_amdgcn_global_load_async_to_lds_b128((v4i*)(src), (v4i*)(dst), 0, 0)
#else
#define WAIT_ASYNC(n) ((void)0)
#define ASYNC_CP16(dst, src) (*(float4*)(dst) = *(const float4*)(src))
#endif

// ---------------------------------------------------------------------------
// Kernel 1: persistent per-batch attention-RNN recurrence.
// One block per batch element (b = blockIdx.x), 256 threads (8 waves).
// Caches key projections K[b,t] = h_t @ W2^T incrementally.
// ---------------------------------------------------------------------------
__global__ __launch_bounds__(256) void rnn_attn_kernel(
    const int* __restrict__ x,        // [B,T]
    const float* __restrict__ embed,  // [V,E]
    const float* __restrict__ W1,     // [H,H]
    const float* __restrict__ W2,     // [H,H]
    const float* __restrict__ vvec,   // [H]
    const float* __restrict__ W_ih,   // [H,E+H]
    const float* __restrict__ b_ih,   // [H]
    const float* __restrict__ W_hh,   // [H,H]
    float* __restrict__ out_hist,     // [B,T,H] workspace (== hist == outs)
    float* __restrict__ Kc,           // [B,T,H] workspace (key cache)
    float* __restrict__ h_fin)        // [B,H]
{
  const int b    = blockIdx.x;
  const int tid  = threadIdx.x;
  const int lane = tid & 31;
  const int wave = tid >> 5;

  __shared__ __align__(16) float sh_h[H_];
  __shared__ __align__(16) float sh_q[H_];
  __shared__ __align__(16) float sh_hn[H_];
  __shared__ __align__(16) float sh_rnn_in[E_ + H_];  // [emb(128) | context(256)]
  __shared__ float sh_sc[T_];
  __shared__ float sh_inv;

  float* outb = out_hist + (size_t)b * T_ * H_;
  float* Kb   = Kc       + (size_t)b * T_ * H_;

  sh_h[tid] = 0.0f;
  __syncthreads();

  for (int t = 0; t < T_; ++t) {
    // ---- phase 1: q = h @ W1^T (thread i -> q[i]) + embedding gather ----
    {
      const float4* w  = (const float4*)(W1 + (size_t)tid * H_);
      const float4* hh = (const float4*)sh_h;
      float a = 0.0f;
      #pragma unroll 8
      for (int k = 0; k < H_ / 4; ++k) {
        float4 wv = w[k], hv = hh[k];
        a = fmaf(wv.x, hv.x, a); a = fmaf(wv.y, hv.y, a);
        a = fmaf(wv.z, hv.z, a); a = fmaf(wv.w, hv.w, a);
      }
      sh_q[tid] = a;
    }
    if (tid < E_) {
      int tok = x[b * T_ + t];
      sh_rnn_in[tid] = embed[(size_t)tok * E_ + tid];
    }
    __syncthreads();

    // ---- phase 2: scores[j] = v . tanh(q + K[j]) for j < t (wave per j) ----
    for (int j = wave; j < t; j += 8) {
      const float* kr = Kb + (size_t)j * H_;
      float s = 0.0f;
      #pragma unroll
      for (int ii = 0; ii < H_ / 32; ++ii) {
        int hx = lane + ii * 32;
        s = fmaf(vvec[hx], tanhf(sh_q[hx] + kr[hx]), s);
      }
      #pragma unroll
      for (int off = 16; off; off >>= 1) s += __shfl_xor(s, off, 32);
      if (lane == 0) sh_sc[j] = s;
    }
    __syncthreads();

    // ---- phase 3: softmax over j < t (single wave) ----
    if (t > 0 && tid < 32) {
      float m = -3.4e38f;
      for (int j = lane; j < t; j += 32) m = fmaxf(m, sh_sc[j]);
      #pragma unroll
      for (int off = 16; off; off >>= 1) m = fmaxf(m, __shfl_xor(m, off, 32));
      float ssum = 0.0f;
      for (int j = lane; j < t; j += 32) {
        float e = __expf(sh_sc[j] - m);
        sh_sc[j] = e;
        ssum += e;
      }
      #pragma unroll
      for (int off = 16; off; off >>= 1) ssum += __shfl_xor(ssum, off, 32);
      if (lane == 0) sh_inv = 1.0f / ssum;
    }
    __syncthreads();

    // ---- phase 4: context[i] = sum_j alpha_j * hist[j,i] ----
    {
      float c = 0.0f;
      for (int j = 0; j < t; ++j) c = fmaf(sh_sc[j], outb[(size_t)j * H_ + tid], c);
      sh_rnn_in[E_ + tid] = (t > 0) ? c * sh_inv : 0.0f;
    }
    __syncthreads();

    // ---- phase 5: h_new = tanh(W_ih @ rnn_in + b_ih + W_hh @ h) ----
    {
      float a = b_ih[tid];
      const float4* wi = (const float4*)(W_ih + (size_t)tid * (E_ + H_));
      const float4* ri = (const float4*)sh_rnn_in;
      #pragma unroll 8
      for (int k = 0; k < (E_ + H_) / 4; ++k) {
        float4 wv = wi[k], rv = ri[k];
        a = fmaf(wv.x, rv.x, a); a = fmaf(wv.y, rv.y, a);
        a = fmaf(wv.z, rv.z, a); a = fmaf(wv.w, rv.w, a);
      }
      const float4* wh = (const float4*)(W_hh + (size_t)tid * H_);
      const float4* hh = (const float4*)sh_h;
      #pragma unroll 8
      for (int k = 0; k < H_ / 4; ++k) {
        float4 wv = wh[k], hv = hh[k];
        a = fmaf(wv.x, hv.x, a); a = fmaf(wv.y, hv.y, a);
        a = fmaf(wv.z, hv.z, a); a = fmaf(wv.w, hv.w, a);
      }
      float hn = tanhf(a);
      sh_hn[tid] = hn;
      outb[(size_t)t * H_ + tid] = hn;
    }
    __syncthreads();

    // ---- phase 6: key cache K[t] = h_new @ W2^T ; rotate h ----
    {
      const float4* w2  = (const float4*)(W2 + (size_t)tid * H_);
      const float4* hn4 = (const float4*)sh_hn;
      float a = 0.0f;
      #pragma unroll 8
      for (int k = 0; k < H_ / 4; ++k) {
        float4 wv = w2[k], hv = hn4[k];
        a = fmaf(wv.x, hv.x, a); a = fmaf(wv.y, hv.y, a);
        a = fmaf(wv.z, hv.z, a); a = fmaf(wv.w, hv.w, a);
      }
      Kb[(size_t)t * H_ + tid] = a;
    }
    if (t == T_ - 1) h_fin[b * H_ + tid] = sh_hn[tid];
    __syncthreads();
    sh_h[tid] = sh_hn[tid];
    __syncthreads();
  }
}

// ---------------------------------------------------------------------------
// Kernel 2: logits = out @ fc_W^T + fc_b via V_WMMA_F32_16X16X4_F32.
// M=4096, K=256, N=32000. Block tile 64(M) x 256(N), 8 waves in a 2x4 grid,
// each wave owns a 32x64 tile = 2x4 WMMA accumulators. K staged in
// double-buffered LDS chunks of 16 via async global->LDS DMA, retired with
// s_wait_asynccnt so the next chunk streams while WMMAs consume the current.
// B tile is n-major in LDS: contiguous 16B per lane for the DMA, and a
// single ds_load_b64 per B fragment. Stride 20 floats = conflict-free.
// ---------------------------------------------------------------------------
#define TM 64
#define TN 256
#define TK 16
#define ALD 20   // padded LDS strides (16B-aligned rows, conflict-free frags)
#define BLD 20
#define NCH (H_ / TK)  // 16 K-chunks

__global__ __launch_bounds__(256) void gemm_logits_kernel(
    const float* __restrict__ A,    // [4096, 256] row-major (out_hist)
    const float* __restrict__ fcW,  // [V, H] row-major -> B[k,n] = fcW[n*H + k]
    const float* __restrict__ fcb,  // [V]
    float* __restrict__ logits)     // [4096, V]
{
  const int tid  = threadIdx.x;
  const int lane = tid & 31;
  const int wave = tid >> 5;
  const int wm   = wave >> 2;        // 0..1 -> m offset wm*32
  const int wn   = wave & 3;         // 0..3 -> n offset wn*64
  const int n0   = blockIdx.x * TN;
  const int m0   = blockIdx.y * TM;

  __shared__ __align__(16) float As[2][TM][ALD];  // 2 * 5120 B
  __shared__ __align__(16) float Bs[2][TN][BLD];  // 2 * 20480 B  (n-major)

  v8f acc[2][4];
  {
    v8f z = {0.f,0.f,0.f,0.f,0.f,0.f,0.f,0.f};
    #pragma unroll
    for (int mi = 0; mi < 2; ++mi)
      #pragma unroll
      for (int nj = 0; nj < 4; ++nj) acc[mi][nj] = z;
  }

  const int a_row = tid >> 2;          // 0..63
  const int a_k4  = (tid & 3) << 2;    // 0,4,8,12
  const int lm    = lane & 15;         // m/n index within 16
  const int kp    = (lane >> 4) << 1;  // 0 or 2 (K-pair select per WMMA layout)

  // Stage one K-chunk (16 wide) into LDS buffer `buf`.
  // A: 64x16 floats = 256 x b128 (1/thread). B: 256x16 = 1024 x b128 (4/thread).
  // -> 5 ASYNCcnt increments per wave per chunk.
  auto stage = [&](int c, int buf) {
    const int kc = c * TK;
    ASYNC_CP16(&As[buf][a_row][a_k4],
               A + (size_t)(m0 + a_row) * H_ + kc + a_k4);
    #pragma unroll
    for (int r = 0; r < 4; ++r) {
      const int i  = tid + 256 * r;
      const int n  = i >> 2;           // 0..255
      const int k4 = (i & 3) << 2;     // 0,4,8,12
      ASYNC_CP16(&Bs[buf][n][k4],
                 fcW + (size_t)(n0 + n) * H_ + kc + k4);
    }
  };

  stage(0, 0);
  for (int c = 0; c < NCH; ++c) {
    const int cur = c & 1;
    if (c + 1 < NCH) {
      stage(c + 1, cur ^ 1);   // overlap next chunk's DMA with this chunk's WMMAs
      WAIT_ASYNC(5);           // this wave's 5 loads for chunk c are complete
    } else {
      WAIT_ASYNC(0);
    }
    __syncthreads();

    // ---- compute: 4 K-substeps of 4, 8 WMMAs each ----
    #pragma unroll
    for (int kk = 0; kk < TK; kk += 4) {
      v2f afr[2], bfr[4];
      afr[0] = *(const v2f*)&As[cur][wm * 32 + lm +  0][kk + kp];
      afr[1] = *(const v2f*)&As[cur][wm * 32 + lm + 16][kk + kp];
      #pragma unroll
      for (int nj = 0; nj < 4; ++nj)
        bfr[nj] = *(const v2f*)&Bs[cur][wn * 64 + nj * 16 + lm][kk + kp];
      #pragma unroll
      for (int mi = 0; mi < 2; ++mi)
        #pragma unroll
        for (int nj = 0; nj < 4; ++nj)
          acc[mi][nj] = WMMA_F32_16x16x4(afr[mi], bfr[nj], acc[mi][nj]);
    }
    __syncthreads();
  }

  // ---- epilogue: D layout (lane<16: n=lane, m=r; lane>=16: n=lane-16, m=8+r) ----
  const int mo = (lane >> 4) * 8;
  #pragma unroll
  for (int mi = 0; mi < 2; ++mi) {
    #pragma unroll
    for (int nj = 0; nj < 4; ++nj) {
      const int mbase = m0 + wm * 32 + mi * 16 + mo;
      const int ncol  = n0 + wn * 64 + nj * 16 + lm;
      const float bias = fcb[ncol];
      #pragma unroll
      for (int r = 0; r < 8; ++r) {
        logits[(size_t)(mbase + r) * V_ + ncol] = acc[mi][nj][r] + bias;
      }
    }
  }
}

// ---------------------------------------------------------------------------
// Launch
// ---------------------------------------------------------------------------
extern "C" void kernel_launch(void* const* d_in, const int* in_sizes, int n_in,
                              void* d_out, int out_size, void* d_ws, size_t ws_size,
                              hipStream_t stream) {
  const int*   x     = (const int*)  d_in[0];
  const float* embed = (const float*)d_in[1];
  const float* W1    = (const float*)d_in[2];
  const float* W2    = (const float*)d_in[3];
  const float* vvec  = (const float*)d_in[4];
  const float* W_ih  = (const float*)d_in[5];
  const float* b_ih  = (const float*)d_in[6];
  const float* W_hh  = (const float*)d_in[7];
  const float* fcW   = (const float*)d_in[8];
  const float* fcb   = (const float*)d_in[9];

  float* logits = (float*)d_out;                              // [B,T,V]
  float* h_fin  = (float*)d_out + (size_t)B_ * T_ * V_;       // [B,H]

  float* out_hist = (float*)d_ws;                             // [B,T,H] fp32
  float* Kcache   = out_hist + (size_t)B_ * T_ * H_;          // [B,T,H] fp32

  // Phase 1: sequential recurrence, batch-parallel (one block per b).
  rnn_attn_kernel<<<dim3(B_), dim3(256), 0, stream>>>(
      x, embed, W1, W2, vvec, W_ih, b_ih, W_hh, out_hist, Kcache, h_fin);

  // Phase 2: big WMMA GEMM -> logits.
  gemm_logits_kernel<<<dim3(V_ / TN, (B_ * T_) / TM), dim3(256), 0, stream>>>(
      out_hist, fcW, fcb, logits);
}